// LDDMMHamilton_31361851195746
// MI455X (gfx1250) — compile-verified
//
#include <hip/hip_runtime.h>

typedef float v2f __attribute__((ext_vector_type(2)));
typedef float v8f __attribute__((ext_vector_type(8)));
typedef unsigned int u32x4 __attribute__((ext_vector_type(4)));
typedef int i32x4 __attribute__((ext_vector_type(4)));
typedef int i32x8 __attribute__((ext_vector_type(8)));

// exp(-d2 * 100) == exp2(d2 * -100*log2(e))
#define EXP2_SCALE (-144.269504088896340736f)

#define CHUNK_PTS   2048
#define CHUNK_ELEMS (CHUNK_PTS * 3)   // 6144 floats = 24 KB per array

__device__ __forceinline__ u32x4 tdm_group0(const float* gptr, unsigned lds_off) {
    unsigned long long ga = (unsigned long long)(uintptr_t)gptr;
    u32x4 g0;
    g0.x = 1u;                                            // count=1, user descriptor
    g0.y = lds_off;                                       // LDS byte address
    g0.z = (unsigned)(ga & 0xFFFFFFFFull);                // global_addr[31:0]
    g0.w = (unsigned)((ga >> 32) & 0x01FFFFFFull)         // global_addr[56:32]
         | (2u << 30);                                    // type = 2 ("image")
    return g0;
}

__global__ __launch_bounds__(256) void lddmm_hamilton_wmma(
    const float* __restrict__ mom,   // p, [N][3]
    const float* __restrict__ q,     // control points, [N][3]
    float* __restrict__ out,         // [0..3N) = -dH/dq, [3N..6N) = dH/dp
    int N)
{
    __shared__ float qs[CHUNK_ELEMS];
    __shared__ float ps[CHUNK_ELEMS];

    const int lane = threadIdx.x & 31;
    const int wave = threadIdx.x >> 5;
    const int n16  = lane & 15;          // matrix column / row-in-tile index
    const int hi   = lane >> 4;          // half-wave id
    const int i0   = (blockIdx.x * 8 + wave) * 16;

    // ---- fixed B operands for this wave's i-tile (B layout: vgpr t holds row t+2*hi, col n16)
    const int ii = (i0 + n16) * 3;
    float qx = q[ii + 0], qy = q[ii + 1], qz = q[ii + 2];
    float px = mom[ii + 0], py = mom[ii + 1], pz = mom[ii + 2];
    const float nqi = qx * qx + qy * qy + qz * qz;   // |q_i|^2, column-constant per lane
    v2f b_qi, b_pi;
    b_qi.x = hi ? qz   : qx;     // rows 0 / 2 of [x,y,z,1]
    b_qi.y = hi ? 1.0f : qy;     // rows 1 / 3
    b_pi.x = hi ? pz   : px;
    b_pi.y = hi ? 0.0f : py;

    v8f acc1 = {};   // cols 0..2: sum_j W_ij q_j ; col 3: rowsum(W)
    v8f acc2 = {};   // cols 0..2: sum_j K_ij p_j

    // TDM descriptor group 1: 4-byte elements, tensor_dim0 = tile_dim0 = 6144,
    // tile_dim1 = 1, tensor_dim0_stride = 6144, no padding / barrier / iterate.
    i32x8 g1;
    g1[0] = 0x00020000;           // data_size = 2 (4 bytes)
    g1[1] = (int)0x18000000;      // tensor_dim0[15:0] = 6144 in bits[63:48]
    g1[2] = 0x00010000;           // tensor_dim1 = 1 in bits[95:80]
    g1[3] = (int)0x18000000;      // tile_dim0 = 6144 in bits[127:112]
    g1[4] = 0x00000001;           // tile_dim1 = 1
    g1[5] = CHUNK_ELEMS;          // tensor_dim0_stride[31:0]
    g1[6] = 0;
    g1[7] = 0;
    const i32x4 gz4 = {};
    const i32x8 gz8 = {};

    const int nchunks = N / CHUNK_PTS;
    for (int chunk = 0; chunk < nchunks; ++chunk) {
        __syncthreads();                          // previous chunk fully consumed
        if (wave == 0) {
            const int base = chunk * CHUNK_ELEMS;
            u32x4 g0q = tdm_group0(q + base,   (unsigned)(uintptr_t)qs);
            u32x4 g0p = tdm_group0(mom + base, (unsigned)(uintptr_t)ps);
            __builtin_amdgcn_tensor_load_to_lds(g0q, g1, gz4, gz4, gz8, 0);
            __builtin_amdgcn_tensor_load_to_lds(g0p, g1, gz4, gz4, gz8, 0);
            __builtin_amdgcn_s_wait_tensorcnt(0);
        }
        __syncthreads();

        for (int j0 = 0; j0 < CHUNK_PTS; j0 += 16) {
            // ---- stage-1 A operands: rows = j-tile points (A layout: vgpr t holds K=t+2*hi)
            const int jj = (j0 + n16) * 3;
            float qj0 = qs[jj + 0], qj1 = qs[jj + 1], qj2 = qs[jj + 2];
            v2f a_qj, a_pj;
            // A_qj rows: [-2x, -2y, -2z, |q_j|^2]
            a_qj.x = hi ? (-2.0f * qj2) : (-2.0f * qj0);
            a_qj.y = hi ? (qj0 * qj0 + qj1 * qj1 + qj2 * qj2) : (-2.0f * qj1);
            a_pj.x = ps[jj + (hi ? 2 : 0)];
            float pyj = ps[jj + 1];
            a_pj.y = hi ? 0.0f : pyj;            // K3 = 0

            v8f zero = {};
            // PPT[j][i] = p_j . p_i
            v8f ppt = __builtin_amdgcn_wmma_f32_16x16x4_f32(
                false, a_pj, false, b_pi, (short)0, zero, false, false);
            // D2R[j][i] = |q_j|^2 - 2 q_j.q_i
            v8f d2r = __builtin_amdgcn_wmma_f32_16x16x4_f32(
                false, a_qj, false, b_qi, (short)0, zero, false, false);

            // ---- K = exp(-d2/sigma^2), W = 400*K*PP  (D layout: vgpr v = row v+8*hi)
            float K[8], W[8];
#pragma unroll
            for (int v = 0; v < 8; ++v) {
                float d2 = d2r[v] + nqi;                       // + |q_i|^2 (lane-constant)
                float k = __builtin_amdgcn_exp2f(d2 * EXP2_SCALE);
                K[v] = k;
                W[v] = 400.0f * k * ppt[v];
            }

            // ---- stage 2: contract over j with 4 chunked 16x16x4 WMMAs per accumulator.
            // A operand of X[i][j] built from the D-layout of the j-row result via
            // uniform half-wave swaps only.
#pragma unroll
            for (int c = 0; c < 4; ++c) {
                v2f aK, aW, bP, bQ;
#pragma unroll
                for (int t = 0; t < 2; ++t) {
                    const int k_lo = 4 * c + t;
                    const int k_hi = k_lo + 2;
                    float xK, xW;
                    if (c < 2) {
                        float swK = __shfl_xor(K[k_hi & 7], 16, 32);
                        float swW = __shfl_xor(W[k_hi & 7], 16, 32);
                        xK = hi ? swK : K[k_lo & 7];
                        xW = hi ? swW : W[k_lo & 7];
                    } else {
                        float swK = __shfl_xor(K[k_lo & 7], 16, 32);
                        float swW = __shfl_xor(W[k_lo & 7], 16, 32);
                        xK = hi ? K[k_hi & 7] : swK;
                        xW = hi ? W[k_hi & 7] : swW;
                    }
                    // B operands: row r = 4c+t+2*hi of [p | 0] and [q | 1]
                    const int r = 4 * c + t + 2 * hi;
                    const int jrow = (j0 + r) * 3;
                    const int nc = (n16 < 3) ? n16 : 0;
                    float pv = ps[jrow + nc];
                    float qv = qs[jrow + nc];
                    float bpv = (n16 < 3) ? pv : 0.0f;
                    float bqv = (n16 < 3) ? qv : ((n16 == 3) ? 1.0f : 0.0f);
                    if (t == 0) { aK.x = xK; aW.x = xW; bP.x = bpv; bQ.x = bqv; }
                    else        { aK.y = xK; aW.y = xW; bP.y = bpv; bQ.y = bqv; }
                }
                acc2 = __builtin_amdgcn_wmma_f32_16x16x4_f32(
                    false, aK, false, bP, (short)0, acc2, false, false);
                acc1 = __builtin_amdgcn_wmma_f32_16x16x4_f32(
                    false, aW, false, bQ, (short)0, acc1, false, false);
            }
        }
    }

    // ---- write outputs. acc[v] at this lane = OUT[row = v+8*hi][col = n16]
    float* out1 = out;                       // -dH/dq
    float* out2 = out + (size_t)N * 3;       // dH/dp
#pragma unroll
    for (int v = 0; v < 8; ++v) {
        float rs = __shfl(acc1[v], 3 + hi * 16, 32);   // rowsum(W) for this row
        if (n16 < 3) {
            const int irow = i0 + v + 8 * hi;
            const float qv = q[irow * 3 + n16];
            out1[irow * 3 + n16] = qv * rs - acc1[v];
            out2[irow * 3 + n16] = 2.0f * acc2[v];
        }
    }
}

extern "C" void kernel_launch(void* const* d_in, const int* in_sizes, int n_in,
                              void* d_out, int out_size, void* d_ws, size_t ws_size,
                              hipStream_t stream) {
    const float* mom = (const float*)d_in[0];            // [1,8192,3]
    const float* cp  = (const float*)d_in[1];            // [1,8192,3]
    float* out = (float*)d_out;                          // 2 * 8192 * 3 floats
    const int N = in_sizes[0] / 3;                       // 8192
    const int blocks = N / 128;                          // 8 waves * 16 rows per block
    lddmm_hamilton_wmma<<<blocks, 256, 0, stream>>>(mom, cp, out, N);
}